// AscendRejectionSampler_6055903888049
// MI455X (gfx1250) — compile-verified
//
#include <hip/hip_runtime.h>
#include <cstdint>
#include <cfloat>

// Speculative-decoding rejection sampler for MI455X (gfx1250, wave32).
//
// Roofline: no matmul -> no WMMA. Bound by streaming target/draft ([N,V] f32
// each ~147MB) + q ([B,V] f32, mostly L2-resident) => ~14us HBM floor at
// 23.3 TB/s. Greedy rows (~half) only need the target stream (argmax), which
// cuts ~45% of total traffic. q rows are staged into LDS by the CDNA5 Tensor
// Data Mover (tensor_load_to_lds + s_wait_tensorcnt), double-buffered, so the
// DMA engine carries one stream while waves issue clause-batched
// global_load_b128 for target/draft (fast path below has no per-load guards
// so the scheduler can keep ~128B/lane in flight).

#define PLACEHOLDER (-1)
#define BLOCK 256
#define CHUNK 4096  // floats per TDM tile (16KB); 2 buffers = 32KB LDS
#define KITER (CHUNK / (BLOCK * 4))  // float4s per thread per chunk

typedef __attribute__((ext_vector_type(4))) unsigned int v4u;
typedef __attribute__((ext_vector_type(8))) unsigned int v8u;

__device__ __forceinline__ unsigned rfl_u32(unsigned x) {
  return (unsigned)__builtin_amdgcn_readfirstlane((int)x);
}

// Issue one TDM descriptor: 1-line tile of `cnt` 4-byte elements,
// global addr {hi,lo} -> LDS byte offset lds_off. 2-SGPR-group form
// (VADDR2/VADDR3 = NULL, tensor up to 2D).
__device__ __forceinline__ void tdm_load_chunk(unsigned lds_off, unsigned lo,
                                               unsigned hi, unsigned cnt) {
  v4u g0;
  g0[0] = 1u;                                // count=1, is_restore=0, no gather
  g0[1] = lds_off;                           // lds_addr (bytes)
  g0[2] = lo;                                // global_addr[31:0]
  g0[3] = (hi & 0x01FFFFFFu) | (2u << 30);   // global_addr[56:32] | type=2
  v8u g1;
  g1[0] = 2u << 16;                          // wg_mask=0, data_size=2 (4B)
  g1[1] = (cnt & 0xFFFFu) << 16;             // tensor_dim0[15:0]
  g1[2] = (cnt >> 16) | (1u << 16);          // tensor_dim0[31:16] | tensor_dim1=1
  g1[3] = (cnt & 0xFFFFu) << 16;             // tile_dim0 (<=65535)
  g1[4] = 1u;                                // tile_dim1=1, tile_dim2=0
  g1[5] = cnt;                               // tensor_dim0_stride[31:0]
  g1[6] = 0u;
  g1[7] = 0u;
  asm volatile("tensor_load_to_lds %0, %1" ::"s"(g0), "s"(g1) : "memory");
}

struct ArgMax {
  float v;
  int i;
};

__device__ __forceinline__ void amax_update(ArgMax& a, float v, int i) {
  if (v > a.v) { a.v = v; a.i = i; }  // ascending i + strict '>' => first max
}

__global__ __launch_bounds__(BLOCK) void rejsam_row_kernel(
    const float* __restrict__ draft, const float* __restrict__ target,
    const float* __restrict__ uni, const float* __restrict__ q,
    const int* __restrict__ cu, const int* __restrict__ did_arr,
    const unsigned char* __restrict__ greedy_arr, int* __restrict__ ws_accept,
    int* __restrict__ ws_token, int B, int V) {
  __shared__ float qbuf[2][CHUNK];
  __shared__ float sTv[BLOCK];
  __shared__ int sTi[BLOCK];
  __shared__ float sRv[BLOCK];
  __shared__ int sRi[BLOCK];

  const int i = blockIdx.x;  // token row
  const int tid = threadIdx.x;

  // req = searchsorted(cu, i, 'right') : first b with cu[b] > i (SALU loop)
  int lo = 0, hi = B - 1;
  while (lo < hi) {
    int mid = (lo + hi) >> 1;
    if (cu[mid] > i) hi = mid; else lo = mid + 1;
  }
  const int req = lo;
  const bool greedy = greedy_arr[req] != 0;

  const float* trow = target + (size_t)i * V;
  const float* drow = draft + (size_t)i * V;
  const uint64_t qrow = (uint64_t)(uintptr_t)(q + (size_t)req * V);

  const bool wave0 = (tid >> 5) == 0;  // wave32: one wave drives the TDM
  const int NC = (V + CHUNK - 1) / CHUNK;

  // Uniform (SGPR) descriptor ingredients.
  const unsigned q_lo = rfl_u32((unsigned)(qrow & 0xFFFFFFFFu));
  const unsigned q_hi = rfl_u32((unsigned)(qrow >> 32));
  // gfx1250 aperture: low 32 bits of a generic LDS pointer == LDS byte offset.
  const unsigned lds0 = rfl_u32((unsigned)(uintptr_t)(&qbuf[0][0]));

  ArgMax bT = {-1.0f, 0};  // target argmax (probs >= 0)
  ArgMax bR = {-1.0f, 0};  // recovered argmax (ratio >= 0)

  if (wave0 && !greedy) {
    unsigned c0 = (unsigned)((V < CHUNK) ? V : CHUNK);
    tdm_load_chunk(lds0, q_lo, q_hi, c0);  // prime buffer 0
  }

  for (int c = 0; c < NC; ++c) {
    const int base = c * CHUNK;
    const int cnt = ((V - base) < CHUNK) ? (V - base) : CHUNK;
    if (wave0 && !greedy) {
      if (c + 1 < NC) {  // prefetch next tile into the other buffer
        const int nb = base + CHUNK;
        const unsigned ncnt = (unsigned)(((V - nb) < CHUNK) ? (V - nb) : CHUNK);
        const uint64_t ga = ((((uint64_t)q_hi) << 32) | q_lo) + (uint64_t)nb * 4u;
        tdm_load_chunk(lds0 + (unsigned)((c + 1) & 1) * (CHUNK * 4u),
                       (unsigned)(ga & 0xFFFFFFFFu), (unsigned)(ga >> 32), ncnt);
        __builtin_amdgcn_s_wait_tensorcnt(1);  // tile c landed (TDM in-order)
      } else {
        __builtin_amdgcn_s_wait_tensorcnt(0);
      }
    }
    __syncthreads();  // tile c visible to all 8 waves
    const float* qc = &qbuf[c & 1][0];

    if (cnt == CHUNK) {
      // ---- fast path: full chunk, no guards -> loads batch into clauses ----
      if (greedy) {
        float4 t4[KITER];
#pragma unroll
        for (int k = 0; k < KITER; ++k)
          t4[k] = *(const float4*)(trow + base + k * (BLOCK * 4) + tid * 4);
#pragma unroll
        for (int k = 0; k < KITER; ++k) {
          const int col = base + k * (BLOCK * 4) + tid * 4;
          amax_update(bT, t4[k].x, col + 0);
          amax_update(bT, t4[k].y, col + 1);
          amax_update(bT, t4[k].z, col + 2);
          amax_update(bT, t4[k].w, col + 3);
        }
      } else {
        float4 t4[KITER], d4[KITER], q4[KITER];
#pragma unroll
        for (int k = 0; k < KITER; ++k) {
          const int off = k * (BLOCK * 4) + tid * 4;
          t4[k] = *(const float4*)(trow + base + off);
          d4[k] = *(const float4*)(drow + base + off);
          q4[k] = *(const float4*)(qc + off);  // ds_load_b128
        }
#pragma unroll
        for (int k = 0; k < KITER; ++k) {
          const int col = base + k * (BLOCK * 4) + tid * 4;
          const float tv[4] = {t4[k].x, t4[k].y, t4[k].z, t4[k].w};
          const float dv[4] = {d4[k].x, d4[k].y, d4[k].z, d4[k].w};
          const float qv[4] = {q4[k].x, q4[k].y, q4[k].z, q4[k].w};
#pragma unroll
          for (int e = 0; e < 4; ++e) {
            amax_update(bT, tv[e], col + e);
            float num = tv[e] - dv[e];
            num = num > 0.0f ? num : 0.0f;
            amax_update(bR, num * __builtin_amdgcn_rcpf(qv[e]), col + e);
          }
        }
      }
    } else {
      // ---- tail path: guarded (cnt is still a multiple of 4) ----
#pragma unroll
      for (int k = 0; k < KITER; ++k) {
        const int off = k * (BLOCK * 4) + tid * 4;
        if (off < cnt) {
          const int col = base + off;
          const float4 tv = *(const float4*)(trow + col);
          const float tvals[4] = {tv.x, tv.y, tv.z, tv.w};
#pragma unroll
          for (int e = 0; e < 4; ++e) amax_update(bT, tvals[e], col + e);
          if (!greedy) {
            const float4 dv = *(const float4*)(drow + col);
            const float4 qv = *(const float4*)(qc + off);
            const float dvals[4] = {dv.x, dv.y, dv.z, dv.w};
            const float qvals[4] = {qv.x, qv.y, qv.z, qv.w};
#pragma unroll
            for (int e = 0; e < 4; ++e) {
              float num = tvals[e] - dvals[e];
              num = num > 0.0f ? num : 0.0f;
              amax_update(bR, num * __builtin_amdgcn_rcpf(qvals[e]), col + e);
            }
          }
        }
      }
    }
    __syncthreads();  // everyone done with buf[c&1] before it is re-filled
  }

  // Block argmax reduction, tie -> lower index (matches jnp.argmax).
  sTv[tid] = bT.v; sTi[tid] = bT.i; sRv[tid] = bR.v; sRi[tid] = bR.i;
  __syncthreads();
  for (int s = BLOCK / 2; s > 0; s >>= 1) {
    if (tid < s) {
      float ov = sTv[tid + s]; int oi = sTi[tid + s];
      if (ov > sTv[tid] || (ov == sTv[tid] && oi < sTi[tid])) { sTv[tid] = ov; sTi[tid] = oi; }
      ov = sRv[tid + s]; oi = sRi[tid + s];
      if (ov > sRv[tid] || (ov == sRv[tid] && oi < sRi[tid])) { sRv[tid] = ov; sRi[tid] = oi; }
    }
    __syncthreads();
  }

  if (tid == 0) {
    const int did = did_arr[i];
    int acc, tok;
    if (greedy) {
      const int targmax = sTi[0];
      acc = (did == targmax);
      tok = targmax;
    } else {
      const float t_at = trow[did];
      const float d_at = drow[did];
      const float u = uni[i];
      const bool rnd = (d_at > 0.0f) && (t_at >= u * d_at);
      acc = rnd ? 1 : 0;
      tok = rnd ? did : sRi[0];
    }
    ws_accept[i] = acc;
    ws_token[i] = tok;
  }
}

// Phase 2: per-request early-exit scan into [B, S+1] int32 output.
__global__ __launch_bounds__(BLOCK) void rejsam_scan_kernel(
    const int* __restrict__ cu, const int* __restrict__ bonus,
    const int* __restrict__ ws_accept, const int* __restrict__ ws_token,
    int* __restrict__ out, int B, int S) {
  const int b = blockIdx.x * blockDim.x + threadIdx.x;
  if (b >= B) return;
  const int start = (b == 0) ? 0 : cu[b - 1];
  const int len = cu[b] - start;
  int* row = out + (size_t)b * (S + 1);
  for (int p = 0; p <= S; ++p) row[p] = PLACEHOLDER;
  bool prefix = true;  // position p written iff all accepts before p
  for (int p = 0; p < len; ++p) {
    if (prefix) row[p] = ws_token[start + p];
    prefix = prefix && (ws_accept[start + p] != 0);
  }
  if (prefix) row[len] = bonus[b];  // bonus only if every draft token accepted
}

extern "C" void kernel_launch(void* const* d_in, const int* in_sizes, int n_in,
                              void* d_out, int out_size, void* d_ws,
                              size_t ws_size, hipStream_t stream) {
  const float* draft = (const float*)d_in[0];            // [N,V] f32
  const float* target = (const float*)d_in[1];           // [N,V] f32
  const float* uni = (const float*)d_in[2];              // [N]   f32
  const float* q = (const float*)d_in[3];                // [B,V] f32
  const int* cu = (const int*)d_in[4];                   // [B]   i32
  const int* did = (const int*)d_in[5];                  // [N]   i32
  const int* bonus = (const int*)d_in[6];                // [B]   i32
  const unsigned char* greedy = (const unsigned char*)d_in[7];  // [B] bool

  const int B = in_sizes[4];
  const int N = in_sizes[5];
  const int V = in_sizes[0] / N;
  const int S = out_size / B - 1;

  int* ws_accept = (int*)d_ws;
  int* ws_token = ws_accept + N;

  rejsam_row_kernel<<<N, BLOCK, 0, stream>>>(draft, target, uni, q, cu, did,
                                             greedy, ws_accept, ws_token, B, V);
  const int blocks2 = (B + BLOCK - 1) / BLOCK;
  rejsam_scan_kernel<<<blocks2, BLOCK, 0, stream>>>(cu, bonus, ws_accept,
                                                    ws_token, (int*)d_out, B, S);
}